// SinglePointEncoder_18056042512601
// MI455X (gfx1250) — compile-verified
//
#include <hip/hip_runtime.h>
#include <hip/hip_bf16.h>

#define EPSF 1e-5f

typedef __attribute__((ext_vector_type(2))) float v2f;
typedef __attribute__((ext_vector_type(8))) float v8f;

// ---------------------------------------------------------------------------
// Farthest point sampling: single workgroup, dist[] in LDS, sequential steps.
// src rows have `stride` floats; first 3 are xyz. Writes idx and gathered xyz.
// ---------------------------------------------------------------------------
__global__ void fps_kernel(const float* __restrict__ src, int stride, int N, int npoint,
                           int* __restrict__ out_idx, float* __restrict__ out_xyz) {
  extern __shared__ float sm[];
  float* dist = sm;                       // N floats
  float* rv = sm + N;                     // blockDim floats
  int* ri = (int*)(rv + blockDim.x);      // blockDim ints
  const int t = threadIdx.x;
  for (int i = t; i < N; i += blockDim.x) dist[i] = 1e10f;
  __syncthreads();

  int far = 0;
  for (int k = 0; k < npoint; ++k) {
    const float cx = src[(size_t)far * stride + 0];
    const float cy = src[(size_t)far * stride + 1];
    const float cz = src[(size_t)far * stride + 2];
    if (t == 0) {
      out_idx[k] = far;
      out_xyz[k * 3 + 0] = cx; out_xyz[k * 3 + 1] = cy; out_xyz[k * 3 + 2] = cz;
    }
    float bv = -1.0f; int bi = 0;
    for (int i = t; i < N; i += blockDim.x) {
      const float dx = src[(size_t)i * stride + 0] - cx;
      const float dy = src[(size_t)i * stride + 1] - cy;
      const float dz = src[(size_t)i * stride + 2] - cz;
      const float d = dx * dx + dy * dy + dz * dz;
      const float dd = fminf(dist[i], d);
      dist[i] = dd;
      if (dd > bv) { bv = dd; bi = i; }   // ascending i keeps first-max
    }
    rv[t] = bv; ri[t] = bi;
    __syncthreads();
    for (int s = blockDim.x >> 1; s > 0; s >>= 1) {
      if (t < s) {
        const float ov = rv[t + s]; const int oi = ri[t + s];
        if (ov > rv[t] || (ov == rv[t] && oi < ri[t])) { rv[t] = ov; ri[t] = oi; }
      }
      __syncthreads();
    }
    far = ri[0];
    __syncthreads();
  }
}

// ---------------------------------------------------------------------------
// Ball query: one wave32 per center; collect the FIRST `32` indices (ascending)
// with d^2 <= r^2; fill remainder with the first hit (or N-1 if no hit).
// ---------------------------------------------------------------------------
__global__ void ball_query_kernel(const float* __restrict__ src, int sstride,
                                  const float* __restrict__ centers,
                                  int N, int S, float r2, int* __restrict__ out) {
  const int wid = (blockIdx.x * blockDim.x + threadIdx.x) >> 5;
  const int lane = threadIdx.x & 31;
  if (wid >= S) return;
  const float cx = centers[wid * 3 + 0];
  const float cy = centers[wid * 3 + 1];
  const float cz = centers[wid * 3 + 2];
  int* row = out + (size_t)wid * 32;
  int cnt = 0;
  int first = N - 1;
  for (int base = 0; base < N && cnt < 32; base += 32) {
    const int j = base + lane;
    bool pred = false;
    if (j < N) {
      const float dx = src[(size_t)j * sstride + 0] - cx;
      const float dy = src[(size_t)j * sstride + 1] - cy;
      const float dz = src[(size_t)j * sstride + 2] - cz;
      pred = (dx * dx + dy * dy + dz * dz) <= r2;
    }
    const unsigned mask = (unsigned)__ballot(pred);
    if (cnt == 0 && mask != 0u) first = base + (__ffs(mask) - 1);
    const int prefix = __popc(mask & ((1u << lane) - 1u));
    if (pred && (cnt + prefix) < 32) row[cnt + prefix] = j;
    cnt += __popc(mask);
  }
  const int filled = cnt < 32 ? cnt : 32;
  if (lane >= filled) row[lane] = first;
}

// SA1 grouping: feat[s*32+j] = {xyz[p]-center, cloud[p](4ch), 0pad} (8 floats)
__global__ void group_sa1_kernel(const float* __restrict__ cloud,
                                 const float* __restrict__ newxyz,
                                 const int* __restrict__ idx,
                                 float* __restrict__ feat, int total) {
  const int gid = blockIdx.x * blockDim.x + threadIdx.x;
  if (gid >= total) return;
  const int s = gid >> 5;
  const int p = idx[gid];
  const float* c = cloud + (size_t)p * 4;
  float* f = feat + (size_t)gid * 8;
  f[0] = c[0] - newxyz[s * 3 + 0];
  f[1] = c[1] - newxyz[s * 3 + 1];
  f[2] = c[2] - newxyz[s * 3 + 2];
  f[3] = c[0]; f[4] = c[1]; f[5] = c[2]; f[6] = c[3]; f[7] = 0.0f;
}

// SA2 grouping: feat[s*32+j] = {l1xyz[p]-center2, l1[p](64ch), 0pad} (68 floats)
__global__ void group_sa2_kernel(const float* __restrict__ l1xyz,
                                 const float* __restrict__ l1,
                                 const float* __restrict__ newxyz2,
                                 const int* __restrict__ idx2,
                                 float* __restrict__ feat, int total) {
  const int gid = blockIdx.x * blockDim.x + threadIdx.x;
  if (gid >= total) return;
  const int s = gid >> 5;
  const int p = idx2[gid];
  float* f = feat + (size_t)gid * 68;
  f[0] = l1xyz[p * 3 + 0] - newxyz2[s * 3 + 0];
  f[1] = l1xyz[p * 3 + 1] - newxyz2[s * 3 + 1];
  f[2] = l1xyz[p * 3 + 2] - newxyz2[s * 3 + 2];
  const float* src = l1 + (size_t)p * 64;
  for (int c = 0; c < 64; ++c) f[3 + c] = src[c];
  f[67] = 0.0f;
}

// Zero-pad weights W[O,C] -> Wp[O,Cp]
__global__ void pad_weight_kernel(const float* __restrict__ W, float* __restrict__ Wp,
                                  int O, int C, int Cp) {
  const int gid = blockIdx.x * blockDim.x + threadIdx.x;
  if (gid >= O * Cp) return;
  const int o = gid / Cp, c = gid % Cp;
  Wp[gid] = (c < C) ? W[o * C + c] : 0.0f;
}

// ---------------------------------------------------------------------------
// WMMA fp32 GEMM: Y[M,O] = act(X[M,K]) @ W[O,K]^T + bias
// act(x) = relu(x*scale+shift) when BN, else identity (compile-time).
// K and O are compile-time so all load/store offsets fold into the 24-bit
// instruction IOFFSET (clause-able immediate-offset stores, minimal addr ALU).
// One wave32 computes FOUR 16x16 M-tiles for one N-tile:
//   - B fragments (and BN scale/shift fragments) loaded once into registers
//   - per M-tile: burst b64 loads of A fragments -> K/4 chained
//     V_WMMA_F32_16X16X4_F32 ops -> immediate-offset store clause.
// Fragment layout per ISA 7.12.2: lane l<16 holds {K=k,k+1}, lane>=16 {k+2,k+3}.
// ---------------------------------------------------------------------------
template<int K, int O, bool BN>
__global__ __launch_bounds__(256)
void wmma_gemm_kernel(const float* __restrict__ X, const float* __restrict__ W,
                      const float* __restrict__ bias, float* __restrict__ Y,
                      int M,
                      const float* __restrict__ in_scale,
                      const float* __restrict__ in_shift) {
  constexpr int KF = K / 4;
  constexpr int tilesN = O / 16;
  const int wave = (blockIdx.x * blockDim.x + threadIdx.x) >> 5;
  const int lane = threadIdx.x & 31;
  const int groups = (M >> 6) * tilesN;   // 4 M-tiles per wave
  if (wave >= groups) return;
  const int tg = wave / tilesN;
  const int tn = wave % tilesN;
  const int half = lane >> 4;
  const int l = lane & 15;
  const int n = tn * 16 + l;
  const float bv = bias[n];

  // B fragments: W[n][k+2*half], W[n][k+2*half+1] for each k step of 4
  v2f bfr[KF];
  const float* wr = W + n * K + 2 * half;
#pragma unroll
  for (int i = 0; i < KF; ++i) bfr[i] = *(const v2f*)(wr + 4 * i);

  v2f sfr[KF], hfr[KF];
  if constexpr (BN) {
    const float* sp = in_scale + 2 * half;
    const float* hp = in_shift + 2 * half;
#pragma unroll
    for (int i = 0; i < KF; ++i) {
      sfr[i] = *(const v2f*)(sp + 4 * i);
      hfr[i] = *(const v2f*)(hp + 4 * i);
    }
  }

#pragma unroll
  for (int t = 0; t < 4; ++t) {
    const int tm = tg * 4 + t;
    const float* xr = X + (size_t)(tm * 16 + l) * K + 2 * half;
    v2f afr[KF];
#pragma unroll
    for (int i = 0; i < KF; ++i) afr[i] = *(const v2f*)(xr + 4 * i);
    if constexpr (BN) {
#pragma unroll
      for (int i = 0; i < KF; ++i) {
        afr[i].x = fmaxf(afr[i].x * sfr[i].x + hfr[i].x, 0.0f);
        afr[i].y = fmaxf(afr[i].y * sfr[i].y + hfr[i].y, 0.0f);
      }
    }
    v8f acc = { bv, bv, bv, bv, bv, bv, bv, bv };
#pragma unroll
    for (int i = 0; i < KF; ++i)
      acc = __builtin_amdgcn_wmma_f32_16x16x4_f32(false, afr[i], false, bfr[i],
                                                  (short)0, acc, false, false);
    // D layout: VGPR v -> row (tm*16 + v + 8*half), col n. O constexpr =>
    // all 8 stores share one base with immediate offsets v*O*4.
    float* yr = Y + (size_t)(tm * 16 + 8 * half) * O + tn * 16 + l;
#pragma unroll
    for (int v = 0; v < 8; ++v) yr[v * O] = acc[v];
  }
}

// Per-channel BN stats (deterministic tree reduce): scale = g*rsqrt(var+eps),
// shift = beta - mean*scale. One block per channel.
__global__ void stats_kernel(const float* __restrict__ Y, int M, int O,
                             const float* __restrict__ g, const float* __restrict__ beta,
                             float* __restrict__ scale, float* __restrict__ shift) {
  const int c = blockIdx.x;
  __shared__ float ss[256], sq[256];
  float s = 0.0f, q = 0.0f;
  for (int r = threadIdx.x; r < M; r += blockDim.x) {
    const float v = Y[(size_t)r * O + c];
    s += v; q += v * v;
  }
  ss[threadIdx.x] = s; sq[threadIdx.x] = q;
  __syncthreads();
  for (int st = 128; st > 0; st >>= 1) {
    if ((int)threadIdx.x < st) { ss[threadIdx.x] += ss[threadIdx.x + st]; sq[threadIdx.x] += sq[threadIdx.x + st]; }
    __syncthreads();
  }
  if (threadIdx.x == 0) {
    const float mean = ss[0] / (float)M;
    const float var = sq[0] / (float)M - mean * mean;
    const float sc = g[c] * rsqrtf(var + EPSF);
    scale[c] = sc;
    shift[c] = beta[c] - mean * sc;
  }
}

// Apply BN+relu then max over nsample=32: L[s,c] = max_j relu(bn(Y[s*32+j, c]))
__global__ void bn_relu_maxpool_kernel(const float* __restrict__ Y,
                                       const float* __restrict__ scale,
                                       const float* __restrict__ shift,
                                       int S, int O, float* __restrict__ L) {
  const int gid = blockIdx.x * blockDim.x + threadIdx.x;
  if (gid >= S * O) return;
  const int s = gid / O, c = gid % O;
  const float sc = scale[c], sh = shift[c];
  const float* y = Y + (size_t)s * 32 * O + c;
  float best = 0.0f;  // relu output is >= 0
  for (int j = 0; j < 32; ++j) {
    const float v = fmaxf(y[(size_t)j * O] * sc + sh, 0.0f);
    best = fmaxf(best, v);
  }
  L[gid] = best;
}

// ---------------------------------------------------------------------------
extern "C" void kernel_launch(void* const* d_in, const int* in_sizes, int n_in,
                              void* d_out, int out_size, void* d_ws, size_t ws_size,
                              hipStream_t stream) {
  (void)n_in; (void)out_size; (void)ws_size;
  const float* cloud = (const float*)d_in[0];
  const float* W[8]; const float* B[8]; const float* G[7]; const float* BE[7];
  auto f = [&](int i) { return (const float*)d_in[i]; };
  if (in_sizes[1] == 112) {  // insertion-order flattening
    W[0]=f(1);  W[1]=f(2);  W[2]=f(3);  B[0]=f(4);  B[1]=f(5);  B[2]=f(6);
    G[0]=f(7);  G[1]=f(8);  G[2]=f(9);  BE[0]=f(10); BE[1]=f(11); BE[2]=f(12);
    W[3]=f(13); W[4]=f(14); W[5]=f(15); B[3]=f(16); B[4]=f(17); B[5]=f(18);
    G[3]=f(19); G[4]=f(20); G[5]=f(21); BE[3]=f(22); BE[4]=f(23); BE[5]=f(24);
    W[6]=f(25); B[6]=f(26); G[6]=f(27); BE[6]=f(28); W[7]=f(29); B[7]=f(30);
  } else {                   // jax tree_leaves (sorted dict keys): proj, sa1, sa2
    B[6]=f(1);  B[7]=f(2);  BE[6]=f(3); G[6]=f(4);  W[6]=f(5);  W[7]=f(6);
    W[0]=f(7);  W[1]=f(8);  W[2]=f(9);  B[0]=f(10); B[1]=f(11); B[2]=f(12);
    G[0]=f(13); G[1]=f(14); G[2]=f(15); BE[0]=f(16); BE[1]=f(17); BE[2]=f(18);
    W[3]=f(19); W[4]=f(20); W[5]=f(21); B[3]=f(22); B[4]=f(23); B[5]=f(24);
    G[3]=f(25); G[4]=f(26); G[5]=f(27); BE[3]=f(28); BE[4]=f(29); BE[5]=f(30);
  }

  const int N1 = 32768, S1 = 2048, S2 = 1024;
  char* ws = (char*)d_ws;
  size_t o = 0;
  auto alloc = [&](size_t bytes) { size_t r = o; o = (o + bytes + 255) & ~(size_t)255; return r; };
  int*   fps1   = (int*)  (ws + alloc(S1 * 4));
  float* nx1    = (float*)(ws + alloc(S1 * 3 * 4));
  int*   idx1   = (int*)  (ws + alloc((size_t)S1 * 32 * 4));
  float* l1     = (float*)(ws + alloc((size_t)S1 * 64 * 4));
  int*   fps2   = (int*)  (ws + alloc(S2 * 4));
  float* nx2    = (float*)(ws + alloc(S2 * 3 * 4));
  int*   idx2   = (int*)  (ws + alloc((size_t)S2 * 32 * 4));
  float* l2     = (float*)(ws + alloc((size_t)S2 * 256 * 4));
  float* p0     = (float*)(ws + alloc((size_t)S2 * 128 * 4));
  float* scaleA = (float*)(ws + alloc(256 * 4));
  float* shiftA = (float*)(ws + alloc(256 * 4));
  float* wp1    = (float*)(ws + alloc(16 * 8 * 4));     // sa1 W0 padded 16x8
  float* wp2    = (float*)(ws + alloc(64 * 68 * 4));    // sa2 W0 padded 64x68
  char*  big    = ws + alloc(0);
  // Phase A (SA1)
  float* feat1 = (float*)(big);
  float* y1    = (float*)(big + 2097152);
  float* y2    = (float*)(big + 2097152 + 4194304);
  float* y3    = (float*)(big + 2097152 + 8388608);
  // Phase B (SA2) — reuses phase-A region
  float* feat2 = (float*)(big);
  float* z1    = (float*)(big + 8912896);
  float* z2    = (float*)(big + 8912896 + 8388608);
  float* z3    = (float*)(big + 8912896 + 16777216);

  auto grid_for = [](int M, int O2) { return (((M >> 6) * (O2 >> 4)) + 7) / 8; };

  // ---- padded weight copies ----
  pad_weight_kernel<<<1, 256, 0, stream>>>(W[0], wp1, 16, 7, 8);
  pad_weight_kernel<<<17, 256, 0, stream>>>(W[3], wp2, 64, 67, 68);

  // ---- SA1 ----
  fps_kernel<<<1, 1024, N1 * 4 + 1024 * 8, stream>>>(cloud, 4, N1, S1, fps1, nx1);
  ball_query_kernel<<<(S1 * 32) / 256, 256, 0, stream>>>(cloud, 4, nx1, N1, S1, 0.16f, idx1);
  group_sa1_kernel<<<(S1 * 32) / 256, 256, 0, stream>>>(cloud, nx1, idx1, feat1, S1 * 32);
  wmma_gemm_kernel<8, 16, false><<<grid_for(65536, 16), 256, 0, stream>>>(
      feat1, wp1, B[0], y1, 65536, nullptr, nullptr);
  stats_kernel<<<16, 256, 0, stream>>>(y1, 65536, 16, G[0], BE[0], scaleA, shiftA);
  wmma_gemm_kernel<16, 16, true><<<grid_for(65536, 16), 256, 0, stream>>>(
      y1, W[1], B[1], y2, 65536, scaleA, shiftA);
  stats_kernel<<<16, 256, 0, stream>>>(y2, 65536, 16, G[1], BE[1], scaleA, shiftA);
  wmma_gemm_kernel<16, 64, true><<<grid_for(65536, 64), 256, 0, stream>>>(
      y2, W[2], B[2], y3, 65536, scaleA, shiftA);
  stats_kernel<<<64, 256, 0, stream>>>(y3, 65536, 64, G[2], BE[2], scaleA, shiftA);
  bn_relu_maxpool_kernel<<<(S1 * 64) / 256, 256, 0, stream>>>(y3, scaleA, shiftA, S1, 64, l1);

  // ---- SA2 ----
  fps_kernel<<<1, 1024, S1 * 4 + 1024 * 8, stream>>>(nx1, 3, S1, S2, fps2, nx2);
  ball_query_kernel<<<(S2 * 32) / 256, 256, 0, stream>>>(nx1, 3, nx2, S1, S2, 0.64f, idx2);
  group_sa2_kernel<<<(S2 * 32) / 256, 256, 0, stream>>>(nx1, l1, nx2, idx2, feat2, S2 * 32);
  wmma_gemm_kernel<68, 64, false><<<grid_for(32768, 64), 256, 0, stream>>>(
      feat2, wp2, B[3], z1, 32768, nullptr, nullptr);
  stats_kernel<<<64, 256, 0, stream>>>(z1, 32768, 64, G[3], BE[3], scaleA, shiftA);
  wmma_gemm_kernel<64, 64, true><<<grid_for(32768, 64), 256, 0, stream>>>(
      z1, W[4], B[4], z2, 32768, scaleA, shiftA);
  stats_kernel<<<64, 256, 0, stream>>>(z2, 32768, 64, G[4], BE[4], scaleA, shiftA);
  wmma_gemm_kernel<64, 256, true><<<grid_for(32768, 256), 256, 0, stream>>>(
      z2, W[5], B[5], z3, 32768, scaleA, shiftA);
  stats_kernel<<<256, 256, 0, stream>>>(z3, 32768, 256, G[5], BE[5], scaleA, shiftA);
  bn_relu_maxpool_kernel<<<(S2 * 256) / 256, 256, 0, stream>>>(z3, scaleA, shiftA, S2, 256, l2);

  // ---- projection head ----
  wmma_gemm_kernel<256, 128, false><<<grid_for(1024, 128), 256, 0, stream>>>(
      l2, W[6], B[6], p0, 1024, nullptr, nullptr);
  stats_kernel<<<128, 256, 0, stream>>>(p0, 1024, 128, G[6], BE[6], scaleA, shiftA);
  wmma_gemm_kernel<128, 16, true><<<grid_for(1024, 16), 256, 0, stream>>>(
      p0, W[7], B[7], (float*)d_out, 1024, scaleA, shiftA);
}